// Encoder_46136538694065
// MI455X (gfx1250) — compile-verified
//
#include <hip/hip_runtime.h>
#include <hip/hip_bf16.h>

#define N_NODES  100000
#define N_EDGES  1600000
#define HID      128
#define N_LAYERS 3
#define N_GRAPHS 500

typedef __attribute__((ext_vector_type(2))) float v2f;
typedef __attribute__((ext_vector_type(8))) float v8f;

// ---------------------------------------------------------------------------
// Edge scatter-add: agg[dst] += h[src].  One wave (32 lanes) per edge; each
// lane loads 4 consecutive feats as one b128 and issues 4 f32 atomics.
// Working set (h, agg = 51.2MB each) is L2-resident on the 192MB L2.
// ---------------------------------------------------------------------------
__global__ void __launch_bounds__(256) scatter_add_kernel(
    const float* __restrict__ h, int lda,
    const int* __restrict__ src, const int* __restrict__ dst,
    float* __restrict__ agg)
{
  int gt   = blockIdx.x * 256 + threadIdx.x;
  int e    = gt >> 5;
  int lane = gt & 31;
  if (e >= N_EDGES) return;
  int s = src[e];
  int d = dst[e];
  const float4* hp = (const float4*)(h + (size_t)s * lda);  // rows 16B aligned
  float4 v = hp[lane];
  float* ap = agg + (size_t)d * HID + lane * 4;
  atomicAdd(ap + 0, v.x);
  atomicAdd(ap + 1, v.y);
  atomicAdd(ap + 2, v.z);
  atomicAdd(ap + 3, v.w);
}

// ---------------------------------------------------------------------------
// Fused WMMA GEMM: out[N,128] = f(A) @ W + bias, plus per-column sum/sumsq
// for the following BatchNorm.  Templated on MODE (branchless k-loop):
//   MODE 0: f(a) = (1+eps)*h + agg              (GIN combine)
//   MODE 1: f(a) = relu(a*scale[k] + shift[k])  (folded BN + ReLU)
// Block = 256 threads = 8 waves covering 256 rows: each wave owns TWO 16-row
// strips x all 128 output columns (16 accumulator tiles of
// V_WMMA_F32_16X16X4_F32, 32 k-steps).  Each LDS B fragment feeds 2 WMMAs.
// W is staged into 64KB LDS pre-swizzled into B-fragment order so each WMMA
// B operand is one conflict-free ds_load_b64; per k-step all 8 fragments are
// loaded as a batch so DS latency overlaps the 16 WMMAs.
// ---------------------------------------------------------------------------
template <int MODE>
__global__ void __launch_bounds__(256) gin_wmma_gemm(
    const float* __restrict__ A, int lda,
    const float* __restrict__ agg,      // MODE 0
    const float* __restrict__ eps_p,    // MODE 0 (device scalar)
    const float* __restrict__ scale,    // MODE 1, per input column
    const float* __restrict__ shiftv,   // MODE 1, per input column
    const float* __restrict__ W,        // [128][128], k-major
    const float* __restrict__ bias,     // [128]
    float* __restrict__ out,            // [N][128]
    float* __restrict__ colsum,
    float* __restrict__ colsq)
{
  __shared__ float ldsW[16384];  // exactly 64 KB

  // Stage W as B fragments: frag e = [tile t(8)][kstep ks(32)][lane(32)],
  // each frag = {W[k][n], W[k+1][n]} with n = t*16 + (lane&15),
  // k = ks*4 + 2*(lane>>4).
  for (int e = threadIdx.x; e < 8192; e += 256) {
    int t  = e >> 10;
    int ks = (e >> 5) & 31;
    int ln = e & 31;
    int n  = (t << 4) | (ln & 15);
    int k  = (ks << 2) + ((ln >> 4) << 1);
    ldsW[2 * e]     = W[k * HID + n];
    ldsW[2 * e + 1] = W[k * HID + HID + n];
  }
  __syncthreads();

  const int lane  = threadIdx.x & 31;
  const int wv    = threadIdx.x >> 5;
  const int row0  = blockIdx.x * 256 + (wv << 4);   // strip 0
  const int row1  = row0 + 128;                     // strip 1
  const int m0    = row0 + (lane & 15);
  const int m1    = row1 + (lane & 15);
  const int r0c   = (m0 < N_NODES) ? m0 : (N_NODES - 1);
  const int r1c   = (m1 < N_NODES) ? m1 : (N_NODES - 1);
  const int khalf = (lane >> 4) << 1;               // 0 or 2

  float oneeps = 0.0f;
  if (MODE == 0) oneeps = 1.0f + eps_p[0];

  v8f acc0[8] = {};
  v8f acc1[8] = {};

  const float* arow0 = A   + (size_t)r0c * lda;
  const float* arow1 = A   + (size_t)r1c * lda;
  const float* grow0 = agg + (size_t)r0c * HID;
  const float* grow1 = agg + (size_t)r1c * HID;
  const v2f*   bfrag = (const v2f*)ldsW;

#pragma unroll 2
  for (int ks = 0; ks < 32; ++ks) {
    const int k = (ks << 2) + khalf;
    v2f a0, a1;
    if (MODE == 0) {
      v2f h0 = *(const v2f*)(arow0 + k);  // 8B-aligned: k even, rows 16B-aligned
      v2f h1 = *(const v2f*)(arow1 + k);
      v2f g0 = *(const v2f*)(grow0 + k);
      v2f g1 = *(const v2f*)(grow1 + k);
      a0[0] = fmaf(oneeps, h0[0], g0[0]);
      a0[1] = fmaf(oneeps, h0[1], g0[1]);
      a1[0] = fmaf(oneeps, h1[0], g1[0]);
      a1[1] = fmaf(oneeps, h1[1], g1[1]);
    } else {
      v2f y0 = *(const v2f*)(arow0 + k);
      v2f y1 = *(const v2f*)(arow1 + k);
      v2f sv = *(const v2f*)(scale + k);
      v2f tv = *(const v2f*)(shiftv + k);
      a0[0] = fmaxf(fmaf(y0[0], sv[0], tv[0]), 0.0f);
      a0[1] = fmaxf(fmaf(y0[1], sv[1], tv[1]), 0.0f);
      a1[0] = fmaxf(fmaf(y1[0], sv[0], tv[0]), 0.0f);
      a1[1] = fmaxf(fmaf(y1[1], sv[1], tv[1]), 0.0f);
    }
    // Batch all 8 B-fragment loads, then 16 WMMAs (2 per fragment).
    v2f bfr[8];
#pragma unroll
    for (int t = 0; t < 8; ++t)
      bfr[t] = bfrag[(t << 10) | (ks << 5) | lane];
#pragma unroll
    for (int t = 0; t < 8; ++t) {
      acc0[t] = __builtin_amdgcn_wmma_f32_16x16x4_f32(
          false, a0, false, bfr[t], (short)0, acc0[t], false, false);
      acc1[t] = __builtin_amdgcn_wmma_f32_16x16x4_f32(
          false, a1, false, bfr[t], (short)0, acc1[t], false, false);
    }
  }

  // Epilogue: + bias, store, per-column sum / sumsq for BatchNorm.
  // C/D layout: VGPR v -> M = v (lanes 0-15) / v+8 (lanes 16-31),
  // N = t*16 + (lane&15).  Both strips fold into one atomic pair per column.
  const int mb0 = row0 + ((lane >> 4) << 3);
  const int mb1 = row1 + ((lane >> 4) << 3);
#pragma unroll
  for (int t = 0; t < 8; ++t) {
    int n = (t << 4) | (lane & 15);
    float bn   = bias[n];
    float csum = 0.0f, csq = 0.0f;
#pragma unroll
    for (int v = 0; v < 8; ++v) {
      int m = mb0 + v;
      float val = acc0[t][v] + bn;
      if (m < N_NODES) {
        out[(size_t)m * HID + n] = val;
        csum += val;
        csq  += val * val;
      }
    }
#pragma unroll
    for (int v = 0; v < 8; ++v) {
      int m = mb1 + v;
      float val = acc1[t][v] + bn;
      if (m < N_NODES) {
        out[(size_t)m * HID + n] = val;
        csum += val;
        csq  += val * val;
      }
    }
    csum += __shfl_xor(csum, 16);   // fold the two M-halves (same n)
    csq  += __shfl_xor(csq, 16);
    if (lane < 16) {
      atomicAdd(&colsum[n], csum);
      atomicAdd(&colsq[n],  csq);
    }
  }
}

// ---------------------------------------------------------------------------
// Fold BN stats into scale/shift:  scale = g*rsqrt(var+eps),
// shift = beta - mu*scale.  Biased variance (matches jnp.var / torch BN).
// ---------------------------------------------------------------------------
__global__ void bn_finalize_kernel(
    const float* __restrict__ colsum, const float* __restrict__ colsq,
    const float* __restrict__ gamma,  const float* __restrict__ beta,
    float* __restrict__ scale, float* __restrict__ shiftv)
{
  int n = threadIdx.x;
  const float invN = 1.0f / (float)N_NODES;
  float mu   = colsum[n] * invN;
  float var  = colsq[n] * invN - mu * mu;
  float rstd = rsqrtf(var + 1e-5f);
  float sc   = gamma[n] * rstd;
  scale[n]  = sc;
  shiftv[n] = beta[n] - mu * sc;
}

// ---------------------------------------------------------------------------
// Outer BN + ReLU, write this layer's slice of node_rep (row stride 384).
// One float4 per thread.
// ---------------------------------------------------------------------------
__global__ void __launch_bounds__(256) bn_relu_store_kernel(
    const float* __restrict__ z,
    const float* __restrict__ scale, const float* __restrict__ shiftv,
    float* __restrict__ node_rep_col)
{
  size_t tid = (size_t)blockIdx.x * 256 + threadIdx.x;   // over N*32
  if (tid >= (size_t)N_NODES * (HID / 4)) return;
  int    q = (int)(tid & 31);           // float4 index within row
  size_t i = tid >> 5;                  // node
  const float4* zp = (const float4*)(z + i * HID);
  const float4* sp = (const float4*)scale;
  const float4* tp = (const float4*)shiftv;
  float4 zv = zp[q], sv = sp[q], tv = tp[q];
  float4 r;
  r.x = fmaxf(fmaf(zv.x, sv.x, tv.x), 0.0f);
  r.y = fmaxf(fmaf(zv.y, sv.y, tv.y), 0.0f);
  r.z = fmaxf(fmaf(zv.z, sv.z, tv.z), 0.0f);
  r.w = fmaxf(fmaf(zv.w, sv.w, tv.w), 0.0f);
  ((float4*)(node_rep_col + i * (3 * HID)))[q] = r;
}

// ---------------------------------------------------------------------------
// Graph pooling: graph_rep[batch[i]] += node_rep[i].  One block per node.
// ---------------------------------------------------------------------------
__global__ void __launch_bounds__(384) pool_kernel(
    const float* __restrict__ node_rep,
    const int* __restrict__ batch,
    float* __restrict__ graph_rep)
{
  int i = blockIdx.x;
  int c = threadIdx.x;
  int g = batch[i];
  atomicAdd(&graph_rep[(size_t)g * (3 * HID) + c],
            node_rep[(size_t)i * (3 * HID) + c]);
}

// ---------------------------------------------------------------------------
extern "C" void kernel_launch(void* const* d_in, const int* in_sizes, int n_in,
                              void* d_out, int out_size, void* d_ws, size_t ws_size,
                              hipStream_t stream)
{
  const float* x     = (const float*)d_in[0];
  const int*   src   = (const int*)d_in[1];
  const int*   dst   = (const int*)d_in[2];
  const int*   batch = (const int*)d_in[3];
  const float* eps   = (const float*)d_in[4];
  const float* W1    = (const float*)d_in[5];
  const float* b1    = (const float*)d_in[6];
  const float* g1    = (const float*)d_in[7];
  const float* be1   = (const float*)d_in[8];
  const float* W2    = (const float*)d_in[9];
  const float* b2    = (const float*)d_in[10];
  const float* gO    = (const float*)d_in[11];
  const float* bO    = (const float*)d_in[12];

  float* out       = (float*)d_out;
  float* graph_rep = out;                                   // [500][384]
  float* node_rep  = out + (size_t)N_GRAPHS * 3 * HID;      // [100000][384]

  // Workspace: bufA (agg, then z of GEMM2), bufY (GEMM1 out), BN stats.
  float* bufA  = (float*)d_ws;
  float* bufY  = bufA + (size_t)N_NODES * HID;
  float* stats = bufY + (size_t)N_NODES * HID;
  float* cs1 = stats;        float* cq1 = stats + 128;
  float* cs2 = stats + 256;  float* cq2 = stats + 384;
  float* sc1 = stats + 512;  float* sh1 = stats + 640;
  float* sc2 = stats + 768;  float* sh2 = stats + 896;

  const size_t nh       = (size_t)N_NODES * HID;
  const int gemm_blocks = (N_NODES + 255) / 256;            // 391
  const int scat_blocks = (int)(((size_t)N_EDGES * 32 + 255) / 256);
  const int ew_blocks   = (int)((nh / 4 + 255) / 256);

  for (int L = 0; L < N_LAYERS; ++L) {
    const float* h  = (L == 0) ? x : (node_rep + (size_t)(L - 1) * HID);
    const int    ld = (L == 0) ? HID : 3 * HID;

    hipMemsetAsync(bufA,  0, nh * sizeof(float), stream);   // agg
    hipMemsetAsync(stats, 0, 512 * sizeof(float), stream);  // cs1/cq1/cs2/cq2

    scatter_add_kernel<<<scat_blocks, 256, 0, stream>>>(h, ld, src, dst, bufA);

    // y = ((1+eps)*h + agg) @ W1 + b1, stats1
    gin_wmma_gemm<0><<<gemm_blocks, 256, 0, stream>>>(
        h, ld, bufA, eps + L, sc1, sh1,
        W1 + (size_t)L * HID * HID, b1 + (size_t)L * HID,
        bufY, cs1, cq1);

    bn_finalize_kernel<<<1, 128, 0, stream>>>(
        cs1, cq1, g1 + (size_t)L * HID, be1 + (size_t)L * HID, sc1, sh1);

    // z = relu(bn(y)) @ W2 + b2, stats2   (z reuses bufA)
    gin_wmma_gemm<1><<<gemm_blocks, 256, 0, stream>>>(
        bufY, HID, bufY /*unused*/, eps + L /*unused*/, sc1, sh1,
        W2 + (size_t)L * HID * HID, b2 + (size_t)L * HID,
        bufA, cs2, cq2);

    bn_finalize_kernel<<<1, 128, 0, stream>>>(
        cs2, cq2, gO + (size_t)L * HID, bO + (size_t)L * HID, sc2, sh2);

    bn_relu_store_kernel<<<ew_blocks, 256, 0, stream>>>(
        bufA, sc2, sh2, node_rep + (size_t)L * HID);
  }

  hipMemsetAsync(graph_rep, 0, (size_t)N_GRAPHS * 3 * HID * sizeof(float), stream);
  pool_kernel<<<N_NODES, 384, 0, stream>>>(node_rep, batch, graph_rep);
}